// MultiHeadSparseDotProductAttentionLayer_29085518528616
// MI455X (gfx1250) — compile-verified
//
#include <hip/hip_runtime.h>
#include <hip/hip_bf16.h>
#include <math.h>

// MI455X / gfx1250: wave32, fp32 WMMA (16x16x4) + TDM async tensor load to LDS.

typedef __attribute__((ext_vector_type(2))) float        v2f;
typedef __attribute__((ext_vector_type(8))) float        v8f;
typedef __attribute__((ext_vector_type(4))) unsigned int v4u;
typedef __attribute__((ext_vector_type(4))) int          v4i;
typedef __attribute__((ext_vector_type(8))) int          v8i;

#define NH 8
#define HD 32
#define CLAMP_V 5.0f

// ---------------------------------------------------------------------------
// Kernel 1: P = softmax(S, axis=1).  S:[8,128].  One wave per row.
// ---------------------------------------------------------------------------
__global__ void softmax_S_kernel(const float* __restrict__ S, float* __restrict__ P) {
  int row  = threadIdx.x >> 5;   // 8 waves = 8 rows
  int lane = threadIdx.x & 31;
  const float* Srow = S + row * 128;
  float v[4];
  float m = -INFINITY;
#pragma unroll
  for (int i = 0; i < 4; ++i) { v[i] = Srow[lane * 4 + i]; m = fmaxf(m, v[i]); }
#pragma unroll
  for (int off = 16; off > 0; off >>= 1) m = fmaxf(m, __shfl_xor(m, off, 32));
  float s = 0.f;
#pragma unroll
  for (int i = 0; i < 4; ++i) { v[i] = __expf(v[i] - m); s += v[i]; }
#pragma unroll
  for (int off = 16; off > 0; off >>= 1) s += __shfl_xor(s, off, 32);
  float inv = 1.0f / s;
  float* Prow = P + row * 128;
#pragma unroll
  for (int i = 0; i < 4; ++i) Prow[lane * 4 + i] = v[i] * inv;
}

// ---------------------------------------------------------------------------
// Kernel 2: fused QKV projection with fp32 WMMA (V_WMMA_F32_16X16X4_F32).
// Block = 256 threads = 8 waves, computes 32 rows x (3 matrices x 256 cols).
// Wave w owns 6 of the 48 16-wide column tiles, times 2 M-tiles (16 rows each)
// -> 12 accumulators (96 VGPRs).
//
// The x tile (32 x 256 f32, row stride 256 dwords) is brought into LDS by the
// Tensor Data Mover: one tensor_load_to_lds issued by wave 0, with the TDM's
// native padding (pad_interval=256 dwords, pad_amount=4 dwords) producing a
// 260-float LDS row stride so the 16 A-rows fan across distinct banks
// (conflict-free ds_load_b64), and tensor_dim1 = N-rowBase so the TDM
// zero-fills out-of-range rows of the tail block for us.
// ---------------------------------------------------------------------------
__global__ __launch_bounds__(256) void qkv_wmma_kernel(
    const float* __restrict__ x,
    const float* __restrict__ Wq, const float* __restrict__ bq,
    const float* __restrict__ Wk, const float* __restrict__ bk,
    const float* __restrict__ Wv, const float* __restrict__ bv,
    float* __restrict__ Q, float* __restrict__ K, float* __restrict__ V,
    int N) {
  __shared__ float xs[32 * 260];
  const int tid     = threadIdx.x;
  const int rowBase = blockIdx.x * 32;

  // ---- TDM async tile load (wave 0 only; scalar-branch gated) ----
  if (__builtin_amdgcn_readfirstlane(tid) < 32) {
    unsigned lds_base = (unsigned)(uintptr_t)&xs[0];       // addr[31:0] == LDS offset
    unsigned long long ga =
        (unsigned long long)(uintptr_t)(x + (size_t)rowBase * 256);
    unsigned rowsLeft = (unsigned)(N - rowBase);           // OOB rows -> zero-filled

    // D# group 0: count=1 | lds_addr | global_addr[56:0] | type=2
    v4u g0 = { 1u,
               lds_base,
               (unsigned)(ga & 0xFFFFFFFFu),
               (unsigned)((ga >> 32) & 0x01FFFFFFu) | (2u << 30) };
    // D# group 1: data_size=4B(2), pad_enable, pad_interval=7(256dw),
    // pad_amount=3(4dw); tensor_dim0=256; tensor_dim1=rowsLeft;
    // tile_dim0=256, tile_dim1=32; tensor_dim0_stride=256.
    v8i g1 = { (int)((2u << 16) | (1u << 20) | (7u << 22) | (3u << 25)),
               (int)(256u << 16),                              // tensor_dim0[15:0]
               (int)(rowsLeft << 16),                          // td0 hi | td1 lo
               (int)(((rowsLeft >> 16) & 0xFFFFu) | (256u << 16)), // td1 hi | tile_dim0
               (int)(32u),                                     // tile_dim1 | tile_dim2=0
               (int)(256),                                     // dim0_stride[31:0]
               0, 0 };                                         // stride hi, dim1_stride
    v4i gz4 = { 0, 0, 0, 0 };
    v8i gz8 = { 0, 0, 0, 0, 0, 0, 0, 0 };
    __builtin_amdgcn_tensor_load_to_lds(g0, g1, gz4, gz4, gz8, 0);
    __builtin_amdgcn_s_wait_tensorcnt(0);
  }
  __syncthreads();

  const int wave = tid >> 5;
  const int lane = tid & 31;
  const int lr   = lane & 15;    // row (A) / col (B,C) within 16x16 tile
  const int hi   = lane >> 4;    // lane-half selects K pair / M+8 rows

  const float* Wmat[3] = {Wq, Wk, Wv};
  const float* Bias[3] = {bq, bk, bv};
  float*       Out[3]  = {Q, K, V};

  const v8f zerov = {0.f, 0.f, 0.f, 0.f, 0.f, 0.f, 0.f, 0.f};
  v8f acc[2][6];
#pragma unroll
  for (int mt = 0; mt < 2; ++mt)
#pragma unroll
    for (int t = 0; t < 6; ++t) acc[mt][t] = zerov;

  const float* Wp[6];
  int colIdx[6];
#pragma unroll
  for (int t = 0; t < 6; ++t) {
    int ct    = wave * 6 + t;          // 0..47
    Wp[t]     = Wmat[ct >> 4];
    colIdx[t] = (ct & 15) * 16 + lr;   // global output column for this lane
  }

#pragma unroll 4
  for (int k0 = 0; k0 < 256; k0 += 4) {
    const int kk = k0 + hi * 2;        // A/B K-pair per lane-half
    v2f a0, a1;
    a0.x = xs[lr * 260 + kk];        a0.y = xs[lr * 260 + kk + 1];
    a1.x = xs[(lr + 16) * 260 + kk]; a1.y = xs[(lr + 16) * 260 + kk + 1];
#pragma unroll
    for (int t = 0; t < 6; ++t) {
      v2f b;
      b.x = Wp[t][(size_t)kk * 256 + colIdx[t]];
      b.y = Wp[t][(size_t)(kk + 1) * 256 + colIdx[t]];
      acc[0][t] = __builtin_amdgcn_wmma_f32_16x16x4_f32(
          false, a0, false, b, (short)0, acc[0][t], false, false);
      acc[1][t] = __builtin_amdgcn_wmma_f32_16x16x4_f32(
          false, a1, false, b, (short)0, acc[1][t], false, false);
    }
  }

  // Epilogue: bias add + store.  C layout: reg r, lane -> row r + hi*8, col lr.
#pragma unroll
  for (int t = 0; t < 6; ++t) {
    int ct = wave * 6 + t;
    float bias  = Bias[ct >> 4][colIdx[t]];
    float* outp = Out[ct >> 4];
#pragma unroll
    for (int mt = 0; mt < 2; ++mt) {
#pragma unroll
      for (int r = 0; r < 8; ++r) {
        int row = rowBase + mt * 16 + r + hi * 8;
        if (row < N) outp[(size_t)row * 256 + colIdx[t]] = acc[mt][t][r] + bias;
      }
    }
  }
}

// ---------------------------------------------------------------------------
// Kernel 3: learned rotary, in-place on Q and K.  One thread per (node, head).
// theta[i] (i=0..15) = sum_a ang[n,a] * P[a, 16h+i];  channel dd uses theta[dd/2];
// rotate_half per head: rot[i<16] = -t[2i+1], rot[16+i] = t[2i].
// ---------------------------------------------------------------------------
__global__ __launch_bounds__(256) void rope_kernel(
    const float* __restrict__ ang, const float* __restrict__ P,
    float* __restrict__ Q, float* __restrict__ K, int N) {
  int gid = blockIdx.x * blockDim.x + threadIdx.x;
  if (gid >= N * NH) return;
  int n = gid >> 3;
  int h = gid & 7;

  float a[8];
#pragma unroll
  for (int j = 0; j < 8; ++j) a[j] = ang[n * 8 + j];

  float cs[16], sn[16];
#pragma unroll
  for (int i = 0; i < 16; ++i) {
    float th = 0.f;
    int pc = h * 16 + i;
#pragma unroll
    for (int j = 0; j < 8; ++j) th = fmaf(a[j], P[j * 128 + pc], th);
    __sincosf(th, &sn[i], &cs[i]);
  }

  size_t base = (size_t)n * 256 + h * 32;
  float q[32], k[32];
  {
    const float4* q4 = reinterpret_cast<const float4*>(Q + base);
    const float4* k4 = reinterpret_cast<const float4*>(K + base);
#pragma unroll
    for (int i = 0; i < 8; ++i) {
      float4 vq = q4[i]; q[4*i] = vq.x; q[4*i+1] = vq.y; q[4*i+2] = vq.z; q[4*i+3] = vq.w;
      float4 vk = k4[i]; k[4*i] = vk.x; k[4*i+1] = vk.y; k[4*i+2] = vk.z; k[4*i+3] = vk.w;
    }
  }

  float qo[32], ko[32];
#pragma unroll
  for (int dd = 0; dd < 32; ++dd) {
    float c = cs[dd >> 1], s = sn[dd >> 1];
    float rq = (dd < 16) ? -q[2 * dd + 1] : q[2 * (dd - 16)];
    float rk = (dd < 16) ? -k[2 * dd + 1] : k[2 * (dd - 16)];
    qo[dd] = q[dd] * c + rq * s;
    ko[dd] = k[dd] * c + rk * s;
  }

  {
    float4* q4 = reinterpret_cast<float4*>(Q + base);
    float4* k4 = reinterpret_cast<float4*>(K + base);
#pragma unroll
    for (int i = 0; i < 8; ++i) {
      q4[i] = make_float4(qo[4*i], qo[4*i+1], qo[4*i+2], qo[4*i+3]);
      k4[i] = make_float4(ko[4*i], ko[4*i+1], ko[4*i+2], ko[4*i+3]);
    }
  }
}

// ---------------------------------------------------------------------------
// Kernel 4: single edge pass.  One wave per edge; lane owns 8 contiguous
// channels, head = 4-lane group.  Scores clamped to ±5 so exp(score-5) gives
// the exact softmax numerator up to a per-dst constant that cancels; den[dst]
// is accumulated with hardware f32 atomics and divided out in the finalize
// kernel (normalization commutes with the segment-sum).
// Q+K+V total 154 MB < 192 MB L2, so the random gathers are L2-resident.
// ---------------------------------------------------------------------------
__global__ __launch_bounds__(256) void edge_kernel(
    const int* __restrict__ ei, const float* __restrict__ Q,
    const float* __restrict__ K, const float* __restrict__ V,
    float* __restrict__ den, float* __restrict__ outAcc, int E) {
  int lane = threadIdx.x & 31;
  int e = blockIdx.x * 8 + (threadIdx.x >> 5);
  if (e >= E) return;
  int src = ei[e];
  int dst = ei[E + e];

  size_t kb = (size_t)src * 256 + lane * 8;
  size_t qb = (size_t)dst * 256 + lane * 8;
  float4 k0 = *reinterpret_cast<const float4*>(K + kb);
  float4 k1 = *reinterpret_cast<const float4*>(K + kb + 4);
  float4 q0 = *reinterpret_cast<const float4*>(Q + qb);
  float4 q1 = *reinterpret_cast<const float4*>(Q + qb + 4);

  float part = k0.x * q0.x + k0.y * q0.y + k0.z * q0.z + k0.w * q0.w
             + k1.x * q1.x + k1.y * q1.y + k1.z * q1.z + k1.w * q1.w;
  // reduce over the 4 lanes of this head (butterfly: all lanes get the sum)
  part += __shfl_xor(part, 1, 32);
  part += __shfl_xor(part, 2, 32);

  float score = part * 0.17677669529663687f;           // 1/sqrt(32)
  score = fminf(fmaxf(score, -CLAMP_V), CLAMP_V);
  float ex = __expf(score - CLAMP_V);                  // shifted softmax numerator

  if ((lane & 3) == 0) unsafeAtomicAdd(&den[dst * NH + (lane >> 2)], ex);

  float4 v0 = *reinterpret_cast<const float4*>(V + kb);
  float4 v1 = *reinterpret_cast<const float4*>(V + kb + 4);
  float* ob = outAcc + qb;
  unsafeAtomicAdd(ob + 0, v0.x * ex);
  unsafeAtomicAdd(ob + 1, v0.y * ex);
  unsafeAtomicAdd(ob + 2, v0.z * ex);
  unsafeAtomicAdd(ob + 3, v0.w * ex);
  unsafeAtomicAdd(ob + 4, v1.x * ex);
  unsafeAtomicAdd(ob + 5, v1.y * ex);
  unsafeAtomicAdd(ob + 6, v1.z * ex);
  unsafeAtomicAdd(ob + 7, v1.w * ex);
}

// ---------------------------------------------------------------------------
// Kernel 5: wV[n,h,:] = acc / (den[n,h] + EPS)
// ---------------------------------------------------------------------------
__global__ void finalize_kernel(float* __restrict__ out,
                                const float* __restrict__ den, int total) {
  int i = blockIdx.x * blockDim.x + threadIdx.x;
  if (i >= total) return;
  out[i] = out[i] / (den[i >> 5] + 1e-16f);
}

// ---------------------------------------------------------------------------
extern "C" void kernel_launch(void* const* d_in, const int* in_sizes, int n_in,
                              void* d_out, int out_size, void* d_ws, size_t ws_size,
                              hipStream_t stream) {
  const float* x   = (const float*)d_in[0];
  const float* ang = (const float*)d_in[1];
  const int*   ei  = (const int*)d_in[2];
  const float* Wq  = (const float*)d_in[3];
  const float* bq  = (const float*)d_in[4];
  const float* Wk  = (const float*)d_in[5];
  const float* bk  = (const float*)d_in[6];
  const float* Wv  = (const float*)d_in[7];
  const float* bv  = (const float*)d_in[8];
  const float* S   = (const float*)d_in[9];

  const int N = in_sizes[0] / 256;
  const int E = in_sizes[2] / 2;
  float* out = (float*)d_out;

  // Workspace layout (floats): P[1024] | Q[N*256] | K[N*256] | V[N*256] | den[N*8]
  float* ws  = (float*)d_ws;
  float* P   = ws;
  float* Q   = ws + 1024;
  float* K   = Q + (size_t)N * 256;
  float* V   = K + (size_t)N * 256;
  float* den = V + (size_t)N * 256;

  (void)hipMemsetAsync(d_out, 0, (size_t)out_size * sizeof(float), stream);
  (void)hipMemsetAsync(den, 0, (size_t)N * NH * sizeof(float), stream);

  softmax_S_kernel<<<1, 256, 0, stream>>>(S, P);

  int gemmBlocks = (N + 31) / 32;
  qkv_wmma_kernel<<<gemmBlocks, 256, 0, stream>>>(x, Wq, bq, Wk, bk, Wv, bv,
                                                  Q, K, V, N);

  int ropeBlocks = (N * NH + 255) / 256;
  rope_kernel<<<ropeBlocks, 256, 0, stream>>>(ang, P, Q, K, N);

  int edgeBlocks = (E + 7) / 8;
  edge_kernel<<<edgeBlocks, 256, 0, stream>>>(ei, Q, K, V, den, out, E);

  int total = N * 256;
  finalize_kernel<<<(total + 255) / 256, 256, 0, stream>>>(out, den, total);
}